// EvaGNN_22462678958350
// MI455X (gfx1250) — compile-verified
//
#include <hip/hip_runtime.h>
#include <hip/hip_bf16.h>
#include <math.h>

typedef __attribute__((ext_vector_type(2))) float v2f;
typedef __attribute__((ext_vector_type(8))) float v8f;

#define IN_DIM  128
#define HID     16
#define OUT_DIM 2

// ---------------------------------------------------------------------------
// Degree / normalization:  deg = 1 (self loop) + #in-edges; dinv = deg^{-1/2}
// ---------------------------------------------------------------------------
__global__ void k_init_dinv(float* __restrict__ dinv, int n) {
    int i = blockIdx.x * blockDim.x + threadIdx.x;
    if (i < n) dinv[i] = 1.0f;               // self-loop contribution
}

__global__ void k_deg_edges(const int* __restrict__ dst, float* __restrict__ deg, int nE) {
    int e = blockIdx.x * blockDim.x + threadIdx.x;
    if (e < nE) unsafeAtomicAdd(&deg[dst[e]], 1.0f);   // global_atomic_add_f32, no return
}

__global__ void k_finalize_dinv(float* __restrict__ dinv, int n) {
    int i = blockIdx.x * blockDim.x + threadIdx.x;
    if (i < n) {
        float d = dinv[i];
        dinv[i] = (d > 0.0f) ? rsqrtf(d) : 0.0f;
    }
}

// ---------------------------------------------------------------------------
// Layer-1 dense GEMM:  h[N,16] = x[N,128] @ W1[128,16]  via V_WMMA_F32_16X16X4_F32
// One wave -> one 16-row tile. 4 waves / block (128 threads).
// A fragment (16x4 f32): lane L holds row M=L%16, K pair kb=(L/16)*2 in 2 VGPRs.
// B fragment (4x16 f32): lane L holds col N=L%16, rows kb, kb+1.
// C/D (16x16 f32): VGPR v, lane L -> (M = v + (L/16)*8, N = L%16).
// ---------------------------------------------------------------------------
__global__ __launch_bounds__(128) void k_gemm1_wmma(const float* __restrict__ x,
                                                    const float* __restrict__ W1,
                                                    float* __restrict__ h, int n) {
    __shared__ float sW[IN_DIM * HID];        // 2048 floats, shared by block
    __shared__ float sX[4][16 * 132];         // per-wave X tile, stride 132 avoids bank conflicts

    // stage W1 cooperatively (block-wide)
    for (int t = threadIdx.x; t < IN_DIM * HID; t += blockDim.x) sW[t] = W1[t];

    const int wave = threadIdx.x >> 5;
    const int lane = threadIdx.x & 31;
    const int tile = blockIdx.x * 4 + wave;
    const int row0 = tile * 16;
    const bool active = (row0 < n);           // wave-uniform

    if (active) {
        // stage 16 rows x 128 floats, coalesced float4 loads (lane = float4 column)
        #pragma unroll 4
        for (int i = 0; i < 16; ++i) {
            int r = row0 + i;
            if (r >= n) r = n - 1;
            const float4 v = ((const float4*)(x + (size_t)r * IN_DIM))[lane];
            *(float4*)&sX[wave][i * 132 + lane * 4] = v;
        }
    }
    __syncthreads();
    if (!active) return;

    const int m  = lane & 15;                 // row within tile (A) / col (B, C/D)
    const int kb = (lane >> 4) << 1;          // 0 or 2
    const float* xr = &sX[wave][m * 132];

    v8f acc = {};
    #pragma unroll
    for (int k0 = 0; k0 < IN_DIM; k0 += 4) {
        v2f a, b;
        a.x = xr[k0 + kb];
        a.y = xr[k0 + kb + 1];
        b.x = sW[(k0 + kb)     * HID + m];
        b.y = sW[(k0 + kb + 1) * HID + m];
        acc = __builtin_amdgcn_wmma_f32_16x16x4_f32(false, a, false, b,
                                                    (short)0, acc, false, false);
    }

    // write back D: lane owns column m of rows rbase..rbase+7
    const int rbase = row0 + ((lane >> 4) << 3);
    #pragma unroll
    for (int v = 0; v < 8; ++v) {
        int r = rbase + v;
        if (r < n) h[(size_t)r * HID + m] = acc[v];
    }
}

// ---------------------------------------------------------------------------
// agg1[i,:] = h[i,:] * dinv[i]^2   (self-loop term, also initializes agg1)
// ---------------------------------------------------------------------------
__global__ void k_selfloop16(const float* __restrict__ h, const float* __restrict__ dinv,
                             float* __restrict__ agg, int n) {
    int t = blockIdx.x * blockDim.x + threadIdx.x;
    if (t >= n * HID) return;
    int i = t / HID;
    float di = dinv[i];
    agg[t] = h[t] * (di * di);
}

// ---------------------------------------------------------------------------
// Layer-1 edge scatter: agg1[dst,:] += h[src,:] * dinv[src]*dinv[dst]
// ---------------------------------------------------------------------------
__global__ void k_scatter1(const int* __restrict__ src, const int* __restrict__ dst,
                           const float* __restrict__ dinv, const float* __restrict__ h,
                           float* __restrict__ agg, int nE) {
    int e = blockIdx.x * blockDim.x + threadIdx.x;
    if (e >= nE) return;
    int s = src[e], d = dst[e];
    float nrm = dinv[s] * dinv[d];
    const float4* hr = (const float4*)(h + (size_t)s * HID);
    float* ar = agg + (size_t)d * HID;
    #pragma unroll
    for (int q = 0; q < 4; ++q) {
        float4 v = hr[q];
        unsafeAtomicAdd(ar + q * 4 + 0, v.x * nrm);
        unsafeAtomicAdd(ar + q * 4 + 1, v.y * nrm);
        unsafeAtomicAdd(ar + q * 4 + 2, v.z * nrm);
        unsafeAtomicAdd(ar + q * 4 + 3, v.w * nrm);
    }
}

// ---------------------------------------------------------------------------
// Layer-2 per node: relu(agg1 + b1) @ W2 -> h2[2]; also agg2 self-loop init
// ---------------------------------------------------------------------------
__global__ void k_layer2(const float* __restrict__ agg1, const float* __restrict__ b1,
                         const float* __restrict__ W2, const float* __restrict__ dinv,
                         float* __restrict__ h2, float* __restrict__ agg2, int n) {
    int i = blockIdx.x * blockDim.x + threadIdx.x;
    if (i >= n) return;
    const float* r = agg1 + (size_t)i * HID;
    float z0 = 0.0f, z1 = 0.0f;
    #pragma unroll
    for (int k = 0; k < HID; ++k) {
        float v = fmaxf(r[k] + b1[k], 0.0f);
        z0 = fmaf(v, W2[k * OUT_DIM + 0], z0);
        z1 = fmaf(v, W2[k * OUT_DIM + 1], z1);
    }
    float di = dinv[i], w = di * di;
    h2[i * 2 + 0]   = z0;
    h2[i * 2 + 1]   = z1;
    agg2[i * 2 + 0] = z0 * w;
    agg2[i * 2 + 1] = z1 * w;
}

// ---------------------------------------------------------------------------
// Layer-2 edge scatter (2 channels)
// ---------------------------------------------------------------------------
__global__ void k_scatter2(const int* __restrict__ src, const int* __restrict__ dst,
                           const float* __restrict__ dinv, const float* __restrict__ h2,
                           float* __restrict__ agg2, int nE) {
    int e = blockIdx.x * blockDim.x + threadIdx.x;
    if (e >= nE) return;
    int s = src[e], d = dst[e];
    float nrm = dinv[s] * dinv[d];
    float2 v = ((const float2*)h2)[s];
    unsafeAtomicAdd(&agg2[d * 2 + 0], v.x * nrm);
    unsafeAtomicAdd(&agg2[d * 2 + 1], v.y * nrm);
}

// ---------------------------------------------------------------------------
// bias + log_softmax over 2 classes
// ---------------------------------------------------------------------------
__global__ void k_final(const float* __restrict__ agg2, const float* __restrict__ b2,
                        float* __restrict__ out, int n) {
    int i = blockIdx.x * blockDim.x + threadIdx.x;
    if (i >= n) return;
    float z0 = agg2[i * 2 + 0] + b2[0];
    float z1 = agg2[i * 2 + 1] + b2[1];
    float mx = fmaxf(z0, z1);
    float l  = mx + logf(expf(z0 - mx) + expf(z1 - mx));
    out[i * 2 + 0] = z0 - l;
    out[i * 2 + 1] = z1 - l;
}

// ---------------------------------------------------------------------------
extern "C" void kernel_launch(void* const* d_in, const int* in_sizes, int n_in,
                              void* d_out, int out_size, void* d_ws, size_t ws_size,
                              hipStream_t stream) {
    const float* x  = (const float*)d_in[0];
    const int*   ei = (const int*)d_in[1];      // edge_index, [2, E] flat
    const float* W1 = (const float*)d_in[2];
    const float* b1 = (const float*)d_in[3];
    const float* W2 = (const float*)d_in[4];
    const float* b2 = (const float*)d_in[5];
    float* out = (float*)d_out;

    const int n  = in_sizes[0] / IN_DIM;        // 100000
    const int nE = in_sizes[1] / 2;             // 3200000
    const int* src = ei;
    const int* dst = ei + nE;

    // workspace layout (floats): dinv[n] | h[16n] | agg1[16n]; h2/agg2 reuse h region
    float* ws   = (float*)d_ws;
    float* dinv = ws;
    float* h    = ws + n;
    float* agg1 = ws + (size_t)n * 17;
    float* h2   = h;                 // reused after scatter1 completes
    float* agg2 = h + (size_t)2 * n;

    const int B = 256;
    k_init_dinv    <<<(n  + B - 1) / B, B, 0, stream>>>(dinv, n);
    k_deg_edges    <<<(nE + B - 1) / B, B, 0, stream>>>(dst, dinv, nE);
    k_finalize_dinv<<<(n  + B - 1) / B, B, 0, stream>>>(dinv, n);

    const int tiles   = (n + 15) / 16;
    const int gblocks = (tiles + 3) / 4;        // 4 waves (tiles) per 128-thread block
    k_gemm1_wmma   <<<gblocks, 128, 0, stream>>>(x, W1, h, n);

    k_selfloop16   <<<(n * HID + B - 1) / B, B, 0, stream>>>(h, dinv, agg1, n);
    k_scatter1     <<<(nE + B - 1) / B, B, 0, stream>>>(src, dst, dinv, h, agg1, nE);
    k_layer2       <<<(n  + B - 1) / B, B, 0, stream>>>(agg1, b1, W2, dinv, h2, agg2, n);
    k_scatter2     <<<(nE + B - 1) / B, B, 0, stream>>>(src, dst, dinv, h2, agg2, nE);
    k_final        <<<(n  + B - 1) / B, B, 0, stream>>>(agg2, b2, out, n);
}